// KGCN_21096879358342
// MI455X (gfx1250) — compile-verified
//
#include <hip/hip_runtime.h>
#include <hip/hip_bf16.h>

typedef __attribute__((ext_vector_type(2))) float v2f;
typedef __attribute__((ext_vector_type(8))) float v8f;

#define DIMK   64
#define N_NEI  16
#define H_S    68   // Hs row stride (floats): 4-bank shift/row -> conflict-free column reads, 16B-aligned
#define WB_S   66   // WBs row stride: 2-bank shift/row -> half-wave B reads hit distinct banks
#define V_S    66   // Vs row stride: rows 8 apart differ by 16 banks -> conflict-free C-tile stores

__global__ __launch_bounds__(256)
void kgcn_kernel(const int*   __restrict__ pairs,
                 const int*   __restrict__ adj_entity,
                 const int*   __restrict__ adj_relation,
                 const float* __restrict__ entity_emb,
                 const float* __restrict__ relation_emb,
                 const float* __restrict__ W,
                 const float* __restrict__ bias,
                 float*       __restrict__ out)
{
    __shared__ float Hs[32 * H_S];    // padded H matrix (rows 0..16 used)
    __shared__ float WBs[64 * WB_S];  // WBs[k][n] = W[n][k]  (B operand = W^T)
    __shared__ float Vs[32 * V_S];    // layer-0 outputs
    __shared__ float us[64];          // user embedding
    __shared__ float bs[64];          // bias
    __shared__ float su[32];          // u . relation_emb[j]  (score table)
    __shared__ float ps[17][16];      // softmax probs; row 16 = p0 (1st hop)
    __shared__ float hs[64];          // layer-1 h vector
    __shared__ float red[64];         // final reduction scratch
    __shared__ int   e1s[16];         // first-hop entity ids

    const int t = threadIdx.x;
    const int s = blockIdx.x;
    const int user = pairs[2 * s];
    const int e0   = pairs[2 * s + 1];

    // ---- Phase 1: stage W^T, bias, user emb, first-hop ids ----
    if (t < 16) e1s[t] = adj_entity[e0 * N_NEI + t];
    if (t < 64) { us[t] = entity_emb[user * DIMK + t]; bs[t] = bias[t]; }
    for (int i = t; i < DIMK * DIMK; i += 256) {
        int n = i >> 6, k = i & 63;
        WBs[k * WB_S + n] = W[i];          // transpose W into B layout
    }
    __syncthreads();

    // ---- Phase 2: score table su[j] = u . relation_emb[j] (32 relations only) ----
    if (t < 32) {
        float acc = 0.f;
        for (int d = 0; d < DIMK; ++d) acc += us[d] * relation_emb[t * DIMK + d];
        su[t] = acc;
    }
    __syncthreads();

    // ---- Phase 3: gather raw scores via relation-index lookup ----
    {
        int m = t >> 4, k = t & 15;                       // 16 second-hop rows
        int r = adj_relation[e1s[m] * N_NEI + k];
        ps[m][k] = su[r];
        if (t < 16) ps[16][t] = su[adj_relation[e0 * N_NEI + t]];  // first-hop row
    }
    __syncthreads();

    // ---- Phase 4: 17 softmaxes over 16 neighbors (max-subtracted, like jax) ----
    if (t < 17) {
        float v[16], mx = -1e30f;
        for (int k = 0; k < 16; ++k) { v[k] = ps[t][k]; mx = fmaxf(mx, v[k]); }
        float sum = 0.f;
        for (int k = 0; k < 16; ++k) { v[k] = expf(v[k] - mx); sum += v[k]; }
        float inv = 1.f / sum;
        for (int k = 0; k < 16; ++k) ps[t][k] = v[k] * inv;
    }
    __syncthreads();

    // ---- Phase 5: build H (rows 0..15: 2nd-hop agg, row 16: 1st-hop agg) ----
    {
        int m = t >> 4, d4 = (t & 15) * 4;                // 16 threads per row, float4 each
        float4 a = *(const float4*)&entity_emb[e1s[m] * DIMK + d4];
        const int* adjrow = &adj_entity[e1s[m] * N_NEI];
        #pragma unroll 4
        for (int k = 0; k < 16; ++k) {
            int e2 = adjrow[k];
            float p = ps[m][k];
            float4 v = *(const float4*)&entity_emb[e2 * DIMK + d4];
            a.x += p * v.x; a.y += p * v.y; a.z += p * v.z; a.w += p * v.w;
        }
        *(float4*)&Hs[m * H_S + d4] = a;
    }
    if (t < 16) {                                         // row 16 (first hop)
        int d4 = t * 4;
        float4 a = *(const float4*)&entity_emb[e0 * DIMK + d4];
        #pragma unroll 4
        for (int k = 0; k < 16; ++k) {
            float p = ps[16][k];
            float4 v = *(const float4*)&entity_emb[e1s[k] * DIMK + d4];
            a.x += p * v.x; a.y += p * v.y; a.z += p * v.z; a.w += p * v.w;
        }
        *(float4*)&Hs[16 * H_S + d4] = a;
    }
    for (int i = t; i < 15 * DIMK; i += 256)              // zero pad rows 17..31
        Hs[(17 + (i >> 6)) * H_S + (i & 63)] = 0.f;
    __syncthreads();

    // ---- Phase 6: V = relu(H @ W^T + b) via V_WMMA_F32_16X16X4_F32 ----
    // 8 waves, one 16x16 output tile each; K=64 as 16 chained k=4 WMMAs.
    {
        int wave = t >> 5, lane = t & 31;
        int mt = wave >> 2, nt = wave & 3;
        int l15   = lane & 15;
        int row   = mt * 16 + l15;                 // A: lanes 0-15 & 16-31 both carry M=0..15
        int col   = nt * 16 + l15;                 // B/D: N striped across lanes
        int khalf = (lane < 16) ? 0 : 2;           // A/B: low half-wave K={0,1}, high K={2,3}
        v8f c = {0.f, 0.f, 0.f, 0.f, 0.f, 0.f, 0.f, 0.f};
        #pragma unroll
        for (int kk = 0; kk < 16; ++kk) {
            int kb = kk * 4 + khalf;
            v2f a;  a.x  = Hs[row * H_S + kb];        a.y  = Hs[row * H_S + kb + 1];
            v2f bb; bb.x = WBs[kb * WB_S + col];      bb.y = WBs[(kb + 1) * WB_S + col];
            c = __builtin_amdgcn_wmma_f32_16x16x4_f32(
                    false, a, false, bb, (short)0, c, false, false);
        }
        #pragma unroll
        for (int r = 0; r < 8; ++r) {              // C/D layout: VGPR r = rows M=r / M=r+8
            int M = mt * 16 + r + ((lane < 16) ? 0 : 8);
            float v = c[r] + bs[col];
            Vs[M * V_S + col] = fmaxf(v, 0.f);     // layer 0 -> ReLU
        }
    }
    __syncthreads();

    // ---- Phase 7: layer 1 (same p0 weights), tanh matvec, sigmoid dot ----
    if (t < 64) {
        float h = Vs[16 * V_S + t];
        #pragma unroll 4
        for (int k = 0; k < 16; ++k) h += ps[16][k] * Vs[k * V_S + t];
        hs[t] = h;
    }
    __syncthreads();
    if (t < 64) {
        float z = bs[t];
        for (int d = 0; d < DIMK; ++d) z += WBs[d * WB_S + t] * hs[d];  // W[t][d]*h[d]
        red[t] = us[t] * tanhf(z);
    }
    __syncthreads();
    if (t == 0) {
        float sum = 0.f;
        for (int d = 0; d < DIMK; ++d) sum += red[d];
        out[s] = 1.f / (1.f + expf(-sum));
    }
}

extern "C" void kernel_launch(void* const* d_in, const int* in_sizes, int n_in,
                              void* d_out, int out_size, void* d_ws, size_t ws_size,
                              hipStream_t stream) {
    const int*   pairs        = (const int*)d_in[0];
    const int*   adj_entity   = (const int*)d_in[1];
    const int*   adj_relation = (const int*)d_in[2];
    const float* entity_emb   = (const float*)d_in[3];
    const float* relation_emb = (const float*)d_in[4];
    const float* W            = (const float*)d_in[5];
    const float* b            = (const float*)d_in[6];
    float*       out          = (float*)d_out;

    int batch = in_sizes[0] / 2;   // pairs is (BATCH, 2)
    kgcn_kernel<<<batch, 256, 0, stream>>>(pairs, adj_entity, adj_relation,
                                           entity_emb, relation_emb, W, b, out);
}